// MoERouterGauss_19825569038530
// MI455X (gfx1250) — compile-verified
//
#include <hip/hip_runtime.h>
#include <math.h>
#include <stdint.h>

typedef float v2f __attribute__((ext_vector_type(2)));
typedef float v8f __attribute__((ext_vector_type(8)));

#define NTOK 16384
#define DIM  2048
#define NE   64
#define TOPK 9
#define KC   32   // K-chunk staged in LDS per iteration
#define APAD 36   // padded LDS row stride (floats) to avoid bank conflicts

#if __has_builtin(__builtin_amdgcn_global_load_async_to_lds_b128)
#define HAVE_ASYNC_LDS 1
#else
#define HAVE_ASYNC_LDS 0
#warning "global_load_async_to_lds_b128 builtin not available; using synchronous LDS staging"
#endif

#if __has_builtin(__builtin_amdgcn_s_wait_asynccnt)
#define WAIT_ASYNC0() __builtin_amdgcn_s_wait_asynccnt(0)
#else
#define WAIT_ASYNC0() asm volatile("s_wait_asynccnt 0x0" ::: "memory")
#endif

// Parameter types revealed by the round-2 diagnostic:
//   param 0: '__attribute__((__vector_size__(4 * sizeof(int)))) int __device__ *'
//   (AS1 = __device__ in HIP printing). Mirror with AS3 for the LDS side.
typedef __attribute__((__vector_size__(4 * sizeof(int)))) int v4i32;
typedef __attribute__((address_space(1))) v4i32* as1_v4i;
typedef __attribute__((address_space(3))) v4i32* as3_v4i;

__device__ __forceinline__ void async_copy16(const float* gsrc, float* lds_dst) {
#if HAVE_ASYNC_LDS
    __builtin_amdgcn_global_load_async_to_lds_b128(
        (as1_v4i)(uintptr_t)gsrc,
        (as3_v4i)(uintptr_t)(uint32_t)(uintptr_t)lds_dst,
        0, 0);
#else
    (void)gsrc; (void)lds_dst;
#endif
}

__device__ __forceinline__ float softplus_f(float v) {
    return (v > 20.0f) ? v : log1pf(__expf(v));
}

__global__ void moe_init_tail(float* __restrict__ tail) {
    tail[threadIdx.x] = 0.0f;   // zero importance[64] + load[64]
}

__global__ __launch_bounds__(128)
void moe_router_kernel(const float* __restrict__ x,
                       const float* __restrict__ Wr,
                       const float* __restrict__ br,
                       const float* __restrict__ Wn,
                       const float* __restrict__ bn,
                       const float* __restrict__ neps,
                       const float* __restrict__ gum,
                       float* __restrict__ out_mask,
                       float* __restrict__ out_prob,
                       float* __restrict__ out_imp,
                       float* __restrict__ out_load) {
    __shared__ __align__(16) float sA[2][16 * APAD];
    __shared__ float s_raw[16][NE];
    __shared__ float s_nsp[16][NE];   // pre-softplus, later overwritten with stddev
    __shared__ float s_noisy[16][NE];
    __shared__ float s_stat[16][8];   // mx_n,sum_n,mx_ng,sum_ng,mx_raw,sum_raw,thr
    __shared__ unsigned long long s_hard[16];

    const int tid  = threadIdx.x;
    const int lane = tid & 31;
    const int wave = tid >> 5;        // 0..3 -> column tile n0 = 16*wave
    const int half = lane >> 4;       // K sub-pair selector for WMMA A/B layout
    const int l16  = lane & 15;
    const int n0   = wave * 16;
    const long rowBase = (long)blockIdx.x * 16;

    v8f accR = {};
    v8f accN = {};

    const int ldRow = tid >> 3;        // 0..15
    const int ldCol = (tid & 7) * 4;   // 0..28 step 4
    const float* gA = x + (rowBase + ldRow) * DIM + ldCol;  // this thread's 16B slice
    const int ldsIdx = ldRow * APAD + ldCol;

#if HAVE_ASYNC_LDS
    // ---- Double-buffered GEMM main loop: HBM -> LDS via async DMA ----
    async_copy16(gA, &sA[0][ldsIdx]);
    int buf = 0;
    for (int k0 = 0; k0 < DIM; k0 += KC, buf ^= 1) {
        WAIT_ASYNC0();                 // chunk k0 has landed in LDS (this wave's part)
        __syncthreads();               // everyone's part landed; prev buffer free
        if (k0 + KC < DIM)
            async_copy16(gA + k0 + KC, &sA[buf ^ 1][ldsIdx]);
        #pragma unroll
        for (int kk = 0; kk < KC; kk += 4) {
            v2f a = *(const v2f*)&sA[buf][l16 * APAD + kk + 2 * half];
            const long krow = (long)(k0 + kk + 2 * half);
            v2f bRv, bNv;
            bRv.x = Wr[krow * NE + n0 + l16];
            bRv.y = Wr[(krow + 1) * NE + n0 + l16];
            bNv.x = Wn[krow * NE + n0 + l16];
            bNv.y = Wn[(krow + 1) * NE + n0 + l16];
            accR = __builtin_amdgcn_wmma_f32_16x16x4_f32(false, a, false, bRv,
                                                         (short)0, accR, false, false);
            accN = __builtin_amdgcn_wmma_f32_16x16x4_f32(false, a, false, bNv,
                                                         (short)0, accN, false, false);
        }
    }
#else
    // ---- Fallback: synchronous staging through VGPRs ----
    for (int k0 = 0; k0 < DIM; k0 += KC) {
        const float4 av = *(const float4*)(gA + k0);
        __syncthreads();
        *(float4*)&sA[0][ldsIdx] = av;
        __syncthreads();
        #pragma unroll
        for (int kk = 0; kk < KC; kk += 4) {
            v2f a = *(const v2f*)&sA[0][l16 * APAD + kk + 2 * half];
            const long krow = (long)(k0 + kk + 2 * half);
            v2f bRv, bNv;
            bRv.x = Wr[krow * NE + n0 + l16];
            bRv.y = Wr[(krow + 1) * NE + n0 + l16];
            bNv.x = Wn[krow * NE + n0 + l16];
            bNv.y = Wn[(krow + 1) * NE + n0 + l16];
            accR = __builtin_amdgcn_wmma_f32_16x16x4_f32(false, a, false, bRv,
                                                         (short)0, accR, false, false);
            accN = __builtin_amdgcn_wmma_f32_16x16x4_f32(false, a, false, bNv,
                                                         (short)0, accN, false, false);
        }
    }
#endif

    // C/D layout: VGPR j -> row M = j + 8*half, col N = n0 + (lane&15)
    __syncthreads();
    #pragma unroll
    for (int j = 0; j < 8; ++j) {
        s_raw[j + 8 * half][n0 + l16] = accR[j];
        s_nsp[j + 8 * half][n0 + l16] = accN[j];
    }
    __syncthreads();

    // ---- Phase 1: per-row stats (16 threads, one row each) ----
    if (tid < 16) {
        const int r = tid;
        const long grow = (rowBase + r) * NE;
        float top[TOPK];
        #pragma unroll
        for (int i = 0; i < TOPK; ++i) top[i] = -INFINITY;
        float mx_n = -INFINITY, mx_ng = -INFINITY, mx_raw = -INFINITY;

        for (int e = 0; e < NE; ++e) {
            const float raw = s_raw[r][e] + br[e];
            const float sd  = softplus_f(s_nsp[r][e] + bn[e]) + 0.01f;
            const float ny  = raw + neps[grow + e] * sd;
            s_raw[r][e]   = raw;
            s_nsp[r][e]   = sd;
            s_noisy[r][e] = ny;
            mx_n   = fmaxf(mx_n, ny);
            mx_ng  = fmaxf(mx_ng, ny + gum[grow + e]);
            mx_raw = fmaxf(mx_raw, raw);
            // branch-free sorted insertion (registers only)
            float v = ny;
            #pragma unroll
            for (int i = 0; i < TOPK; ++i) {
                const float t  = top[i];
                const float hi = fmaxf(t, v);
                v = fminf(t, v);
                top[i] = hi;
            }
        }
        const float thr = top[TOPK - 1];

        float sum_n = 0.f, sum_ng = 0.f, sum_raw = 0.f;
        int c_gt = 0;
        for (int e = 0; e < NE; ++e) {
            const float ny = s_noisy[r][e];
            sum_n   += __expf(ny - mx_n);
            sum_ng  += __expf(ny + gum[grow + e] - mx_ng);
            sum_raw += __expf(s_raw[r][e] - mx_raw);
            c_gt += (ny > thr) ? 1 : 0;
        }
        // hard mask with exact-K tie handling (lowest index first, like top_k)
        unsigned long long hm = 0ull;
        int eq_take = TOPK - c_gt;
        for (int e = 0; e < NE; ++e) {
            const float ny = s_noisy[r][e];
            if (ny > thr) {
                hm |= (1ull << e);
            } else if (ny == thr && eq_take > 0) {
                hm |= (1ull << e);
                --eq_take;
            }
        }
        s_hard[r] = hm;
        s_stat[r][0] = mx_n;   s_stat[r][1] = sum_n;
        s_stat[r][2] = mx_ng;  s_stat[r][3] = sum_ng;
        s_stat[r][4] = mx_raw; s_stat[r][5] = sum_raw;
        s_stat[r][6] = thr;
    }
    __syncthreads();

    // ---- Phase 2a: write expert_mask & route_prob (all 128 threads) ----
    {
        const int r  = tid >> 3;
        const int e0 = (tid & 7) * 8;
        const long grow = (rowBase + r) * NE;
        const float mx_n    = s_stat[r][0];
        const float inv_sn  = 1.0f / s_stat[r][1];
        const float mx_ng   = s_stat[r][2];
        const float inv_sng = 1.0f / s_stat[r][3];
        const unsigned long long hm = s_hard[r];
        #pragma unroll
        for (int j = 0; j < 8; ++j) {
            const int e = e0 + j;
            const float ny = s_noisy[r][e];
            out_prob[grow + e] = __expf(ny - mx_n) * inv_sn;
            const float ms   = __expf(ny + gum[grow + e] - mx_ng) * inv_sng;
            const float hard = (float)((hm >> e) & 1ull);
            out_mask[grow + e] = (hard - ms) + ms;   // matches reference rounding
        }
    }

    // ---- Phase 2b: per-block importance/load partials -> one atomic per expert ----
    if (tid < NE) {
        const int e = tid;
        float imp = 0.f, ld = 0.f;
        #pragma unroll
        for (int r = 0; r < 16; ++r) {
            const float raw = s_raw[r][e];
            imp += __expf(raw - s_stat[r][4]) / s_stat[r][5];
            const float z = (s_stat[r][6] - raw) / s_nsp[r][e];
            // 1 - 0.5*(1+erf(z/(SIGMA*sqrt(2)))) ; SIGMA = 1/64 -> scale = 64/sqrt(2)
            ld += 0.5f - 0.5f * erff(z * 45.254834f);
        }
        atomicAdd(&out_imp[e], imp);
        atomicAdd(&out_load[e], ld);
    }
}

extern "C" void kernel_launch(void* const* d_in, const int* in_sizes, int n_in,
                              void* d_out, int out_size, void* d_ws, size_t ws_size,
                              hipStream_t stream) {
    const float* x    = (const float*)d_in[0];
    const float* Wr   = (const float*)d_in[1];
    const float* br   = (const float*)d_in[2];
    const float* Wn   = (const float*)d_in[3];
    const float* bn   = (const float*)d_in[4];
    const float* neps = (const float*)d_in[5];
    const float* gum  = (const float*)d_in[6];

    float* out       = (float*)d_out;
    float* out_mask  = out;                         // N*E
    float* out_prob  = out + (long)NTOK * NE;       // N*E
    float* out_imp   = out + 2L * NTOK * NE;        // E
    float* out_load  = out_imp + NE;                // E

    moe_init_tail<<<1, 2 * NE, 0, stream>>>(out_imp);
    moe_router_kernel<<<NTOK / 16, 128, 0, stream>>>(
        x, Wr, br, Wn, bn, neps, gum, out_mask, out_prob, out_imp, out_load);
}